// MambaChordClassifier_14448269983811
// MI455X (gfx1250) — compile-verified
//
#include <hip/hip_runtime.h>
#include <hip/hip_bf16.h>

// ---------------- problem constants ----------------
#define B_      8
#define L_      2048
#define D_IN    24
#define D_MODEL 128
#define N_CLS   24
#define D_INNER 256
#define D_STATE 16
#define DT_RANK 8
#define D_CONV  4
#define NROWS   (B_ * L_)          // 16384 token rows
#define SEG     32                 // scan segments
#define LSEG    (L_ / SEG)         // 64 steps per segment

// ---------------- workspace layout (float offsets) ----------------
#define WS_WCIN   0u               // [24 x 512]  fused W_in @ in_proj_w
#define WS_BCIN   12288u           // [512]       fused bias
#define WS_WDT    12800u           // [256 x 256] x_proj[:,0:8] @ dt_proj
#define WS_WCOUT  78336u           // [256 x 32]  out_proj @ W_cls (zero-padded)
#define WS_ANEG   86528u           // [256 x 16]  A = -exp(A_log)
#define WS_XZ     90624u           // [16384 x 512]
#define WS_XC     8479232u         // [16384 x 256]  xc, overwritten by y2 in pass2
#define WS_BC     12673536u        // [16384 x 32]   B(16) | C(16)
#define WS_DELTA  13197824u        // [16384 x 256]
#define WS_P      17392128u        // [8 x 32 x 256 x 16] segment products
#define WS_Q      18440704u        // [8 x 32 x 256 x 16] segment end states
#define WS_H0     19489280u        // [8 x 32 x 256 x 16] segment start states

typedef float v2f __attribute__((ext_vector_type(2)));
typedef float v8f __attribute__((ext_vector_type(8)));

__device__ __forceinline__ float fast_sigmoid(float x) {
    return 1.0f / (1.0f + __expf(-x));
}
__device__ __forceinline__ float fast_softplus(float x) {
    return (x > 20.0f) ? x : __logf(1.0f + __expf(x));
}

// ---------------------------------------------------------------------------
// fp32 WMMA tile engine: one wave computes 4 M-tiles x 1 N-tile (64x16 output)
// using V_WMMA_F32_16X16X4_F32.  A: row-major [M,lda], B: row-major [K,ldb].
// A frag (16x4):  lane<16 -> K=k+0/1, lane>=16 -> K=k+2/3, M = lane&15
// B frag (4x16):  vgpr v,lane<16 -> row k+v, lane>=16 -> row k+v+2, N = lane&15
// ---------------------------------------------------------------------------
__device__ __forceinline__ void wmma_tile4_f32(const float* __restrict__ A, int lda,
                                               const float* __restrict__ B, int ldb,
                                               int K, int m0, int bcol0, v8f acc[4])
{
    const int lane = threadIdx.x & 31;
    const int nn   = lane & 15;
    const int kh   = (lane >> 4) << 1;          // 0 or 2
    for (int k = 0; k < K; k += 4) {
        v2f bf;
        bf.x = B[(k + kh + 0) * ldb + bcol0 + nn];
        bf.y = B[(k + kh + 1) * ldb + bcol0 + nn];
#pragma unroll
        for (int mt = 0; mt < 4; ++mt) {
            const int m = m0 + mt * 16 + nn;
            v2f af;
            af.x = A[m * lda + k + kh + 0];
            af.y = A[m * lda + k + kh + 1];
            acc[mt] = __builtin_amdgcn_wmma_f32_16x16x4_f32(
                false, af, false, bf, (short)0, acc[mt], false, false);
        }
    }
}

__device__ __forceinline__ void zero_acc(v8f acc[4]) {
#pragma unroll
    for (int i = 0; i < 4; ++i)
#pragma unroll
        for (int j = 0; j < 8; ++j) acc[i][j] = 0.0f;
}

// ---------------------------------------------------------------------------
// K0: precompute fused weights (tiny; scalar VALU)
// ---------------------------------------------------------------------------
__global__ void precompute_kernel(const float* __restrict__ W_in, const float* __restrict__ b_in,
                                  const float* __restrict__ in_proj, const float* __restrict__ x_proj,
                                  const float* __restrict__ dt_proj, const float* __restrict__ out_proj,
                                  const float* __restrict__ W_cls,  const float* __restrict__ A_log,
                                  float* __restrict__ Wc_in, float* __restrict__ bc_in,
                                  float* __restrict__ Wdt,   float* __restrict__ Wc_out,
                                  float* __restrict__ Aneg)
{
    const int idx = blockIdx.x * 256 + threadIdx.x;
    if (idx < 12288) {                                  // Wc_in[24][512]
        const int i = idx >> 9, j = idx & 511;
        float s = 0.0f;
        for (int k = 0; k < D_MODEL; ++k) s += W_in[i * D_MODEL + k] * in_proj[k * 512 + j];
        Wc_in[idx] = s;
    } else if (idx < 12800) {                           // bc_in[512]
        const int j = idx - 12288;
        float s = 0.0f;
        for (int k = 0; k < D_MODEL; ++k) s += b_in[k] * in_proj[k * 512 + j];
        bc_in[j] = s;
    } else if (idx < 78336) {                           // Wdt[256][256]
        const int t = idx - 12800;
        const int i = t >> 8, j = t & 255;
        float s = 0.0f;
        for (int r = 0; r < DT_RANK; ++r) s += x_proj[i * 40 + r] * dt_proj[r * D_INNER + j];
        Wdt[t] = s;
    } else if (idx < 86528) {                           // Wc_out[256][32] (cols>=24 zero)
        const int t = idx - 78336;
        const int i = t >> 5, j = t & 31;
        float s = 0.0f;
        if (j < N_CLS)
            for (int k = 0; k < D_MODEL; ++k) s += out_proj[i * D_MODEL + k] * W_cls[k * N_CLS + j];
        Wc_out[t] = s;
    } else if (idx < 90624) {                           // Aneg[256*16]
        const int t = idx - 86528;
        Aneg[t] = -__expf(A_log[t]);
    }
}

// ---------------------------------------------------------------------------
// GEMM1: xz[16384,512] = input[16384,24] @ Wc_in[24,512] + bc_in
// ---------------------------------------------------------------------------
__global__ void gemm_in_kernel(const float* __restrict__ A, const float* __restrict__ B,
                               const float* __restrict__ bias, float* __restrict__ C)
{
    const int wave = (blockIdx.x * blockDim.x + threadIdx.x) >> 5;   // 0..8191
    const int mg = wave >> 5;          // 256 M-groups of 64 rows
    const int nt = wave & 31;          // 32 N-tiles
    const int m0 = mg * 64, n0 = nt * 16;
    v8f acc[4]; zero_acc(acc);
    wmma_tile4_f32(A, D_IN, B, 512, D_IN, m0, n0, acc);
    const int lane = threadIdx.x & 31;
    const int col  = n0 + (lane & 15);
    const float bv = bias[col];
#pragma unroll
    for (int mt = 0; mt < 4; ++mt)
#pragma unroll
        for (int r = 0; r < 8; ++r) {
            const int row = m0 + mt * 16 + r + ((lane >> 4) << 3);
            C[row * 512 + col] = acc[mt][r] + bv;
        }
}

// ---------------------------------------------------------------------------
// Depthwise causal conv1d (K=4) + SiLU:  xc[b,l,d] from xm = xz[...,0:256]
// ---------------------------------------------------------------------------
__global__ void conv_silu_kernel(const float* __restrict__ xz, const float* __restrict__ cw,
                                 const float* __restrict__ cb, float* __restrict__ xc)
{
    const int bl = blockIdx.x;          // b*L + l
    const int d  = threadIdx.x;         // 0..255
    const int l  = bl & (L_ - 1);
    float acc = cb[d];
#pragma unroll
    for (int k = 0; k < D_CONV; ++k) {
        const int ll = l - (D_CONV - 1) + k;
        if (ll >= 0) acc += cw[d * D_CONV + k] * xz[(bl - (D_CONV - 1) + k) * 512 + d];
    }
    xc[bl * D_INNER + d] = acc * fast_sigmoid(acc);     // silu
}

// ---------------------------------------------------------------------------
// GEMM2: [16384,256] @ [256, 32|256] -> BC (cols 0..31) and delta (softplus)
// ---------------------------------------------------------------------------
__global__ void gemm_xproj_kernel(const float* __restrict__ xc, const float* __restrict__ xpw,
                                  const float* __restrict__ Wdt, const float* __restrict__ dtb,
                                  float* __restrict__ BC, float* __restrict__ delta)
{
    const int wave = (blockIdx.x * blockDim.x + threadIdx.x) >> 5;   // 0..4607
    const int mg = wave / 18;
    const int nt = wave % 18;          // 18 N-tiles: 2 for B/C, 16 for delta
    const int m0 = mg * 64, n0 = nt * 16;
    const float* Bp; int ldb, bcol;
    if (n0 < 32) { Bp = xpw + DT_RANK; ldb = 40;      bcol = n0; }        // x_proj B/C cols
    else         { Bp = Wdt;           ldb = D_INNER; bcol = n0 - 32; }   // fused dt weight
    v8f acc[4]; zero_acc(acc);
    wmma_tile4_f32(xc, D_INNER, Bp, ldb, D_INNER, m0, bcol, acc);
    const int lane = threadIdx.x & 15 ? (threadIdx.x & 31) : (threadIdx.x & 31);
    const int ln = threadIdx.x & 31;
    const int nn = ln & 15;
    (void)lane;
    if (n0 < 32) {
        const int col = n0 + nn;
#pragma unroll
        for (int mt = 0; mt < 4; ++mt)
#pragma unroll
            for (int r = 0; r < 8; ++r) {
                const int row = m0 + mt * 16 + r + ((ln >> 4) << 3);
                BC[row * 32 + col] = acc[mt][r];
            }
    } else {
        const int dcol = n0 - 32 + nn;
        const float bv = dtb[dcol];
#pragma unroll
        for (int mt = 0; mt < 4; ++mt)
#pragma unroll
            for (int r = 0; r < 8; ++r) {
                const int row = m0 + mt * 16 + r + ((ln >> 4) << 3);
                delta[row * D_INNER + dcol] = fast_softplus(acc[mt][r] + bv);
            }
    }
}

// ---------------------------------------------------------------------------
// Segmented linear scan, pass 1: per-segment (prod a, end-state q)
//   h_l = a_l * h_{l-1} + u_l ;  a = exp(delta*A_s), u = delta*x*B_s
// ---------------------------------------------------------------------------
__global__ void scan_pass1_kernel(const float* __restrict__ delta, const float* __restrict__ xc,
                                  const float* __restrict__ BC, const float* __restrict__ Aneg,
                                  float* __restrict__ P, float* __restrict__ Q)
{
    const int seg = blockIdx.x, b = blockIdx.y, d = threadIdx.x;
    __shared__ float sBC[256];
    float Ar[D_STATE], h[D_STATE], Pp[D_STATE];
#pragma unroll
    for (int s = 0; s < D_STATE; ++s) { Ar[s] = Aneg[d * D_STATE + s]; h[s] = 0.0f; Pp[s] = 1.0f; }
    const int l0 = seg * LSEG;
    for (int c = 0; c < LSEG / 8; ++c) {
        __syncthreads();
        sBC[threadIdx.x] = BC[(b * L_ + l0 + c * 8) * 32 + threadIdx.x];
        __syncthreads();
#pragma unroll
        for (int t = 0; t < 8; ++t) {
            const int idx = (b * L_ + l0 + c * 8 + t) * D_INNER + d;
            const float dl = delta[idx];
            const float dx = dl * xc[idx];
#pragma unroll
            for (int s = 0; s < D_STATE; ++s) {
                const float a = __expf(dl * Ar[s]);
                h[s] = a * h[s] + dx * sBC[t * 32 + s];
                Pp[s] *= a;
            }
        }
    }
    const int base = ((b * SEG + seg) * D_INNER + d) * D_STATE;
#pragma unroll
    for (int s = 0; s < D_STATE; ++s) { P[base + s] = Pp[s]; Q[base + s] = h[s]; }
}

// pass mid: chain the 32 segment carries per (b,d,s)
__global__ void scan_mid_kernel(const float* __restrict__ P, const float* __restrict__ Q,
                                float* __restrict__ H0)
{
    const int b = blockIdx.x, d = threadIdx.x;
    float carry[D_STATE];
#pragma unroll
    for (int s = 0; s < D_STATE; ++s) carry[s] = 0.0f;
    for (int seg = 0; seg < SEG; ++seg) {
        const int base = ((b * SEG + seg) * D_INNER + d) * D_STATE;
#pragma unroll
        for (int s = 0; s < D_STATE; ++s) {
            H0[base + s] = carry[s];
            carry[s] = P[base + s] * carry[s] + Q[base + s];
        }
    }
}

// pass 2: replay each segment from true h0, emit y2 = (y + xc*D)*silu(z)
// (written in place over xc)
__global__ void scan_pass2_kernel(float* __restrict__ xcy, const float* __restrict__ delta,
                                  const float* __restrict__ BC, const float* __restrict__ Aneg,
                                  const float* __restrict__ H0, const float* __restrict__ Dp,
                                  const float* __restrict__ xz)
{
    const int seg = blockIdx.x, b = blockIdx.y, d = threadIdx.x;
    __shared__ float sBC[256];
    float Ar[D_STATE], h[D_STATE];
    const int base = ((b * SEG + seg) * D_INNER + d) * D_STATE;
#pragma unroll
    for (int s = 0; s < D_STATE; ++s) { Ar[s] = Aneg[d * D_STATE + s]; h[s] = H0[base + s]; }
    const float Dd = Dp[d];
    const int l0 = seg * LSEG;
    for (int c = 0; c < LSEG / 8; ++c) {
        __syncthreads();
        sBC[threadIdx.x] = BC[(b * L_ + l0 + c * 8) * 32 + threadIdx.x];
        __syncthreads();
#pragma unroll
        for (int t = 0; t < 8; ++t) {
            const int bl  = b * L_ + l0 + c * 8 + t;
            const int idx = bl * D_INNER + d;
            const float dl = delta[idx];
            const float xv = xcy[idx];
            const float dx = dl * xv;
            float y = 0.0f;
#pragma unroll
            for (int s = 0; s < D_STATE; ++s) {
                const float a = __expf(dl * Ar[s]);
                h[s] = a * h[s] + dx * sBC[t * 32 + s];
                y += h[s] * sBC[t * 32 + 16 + s];
            }
            const float zv = xz[bl * 512 + D_INNER + d];
            xcy[idx] = (y + xv * Dd) * zv * fast_sigmoid(zv);
        }
    }
}

// ---------------------------------------------------------------------------
// GEMM4: logits[16384,24] = y2[16384,256] @ Wc_out[256,32] + b_cls (cols<24)
// ---------------------------------------------------------------------------
__global__ void gemm_out_kernel(const float* __restrict__ y2, const float* __restrict__ Wco,
                                const float* __restrict__ bcls, float* __restrict__ out)
{
    const int wave = (blockIdx.x * blockDim.x + threadIdx.x) >> 5;   // 0..511
    const int mg = wave >> 1;
    const int nt = wave & 1;
    const int m0 = mg * 64, n0 = nt * 16;
    v8f acc[4]; zero_acc(acc);
    wmma_tile4_f32(y2, D_INNER, Wco, 32, D_INNER, m0, n0, acc);
    const int ln = threadIdx.x & 31;
    const int col = n0 + (ln & 15);
    if (col < N_CLS) {
        const float bv = bcls[col];
#pragma unroll
        for (int mt = 0; mt < 4; ++mt)
#pragma unroll
            for (int r = 0; r < 8; ++r) {
                const int row = m0 + mt * 16 + r + ((ln >> 4) << 3);
                out[row * N_CLS + col] = acc[mt][r] + bv;
            }
    }
}

// ---------------------------------------------------------------------------
extern "C" void kernel_launch(void* const* d_in, const int* in_sizes, int n_in,
                              void* d_out, int out_size, void* d_ws, size_t ws_size,
                              hipStream_t stream)
{
    (void)in_sizes; (void)n_in; (void)out_size; (void)ws_size;
    const float* input    = (const float*)d_in[0];
    const float* W_in     = (const float*)d_in[1];
    const float* b_in     = (const float*)d_in[2];
    const float* in_proj  = (const float*)d_in[3];
    const float* conv_w   = (const float*)d_in[4];
    const float* conv_b   = (const float*)d_in[5];
    const float* x_proj   = (const float*)d_in[6];
    const float* dt_proj  = (const float*)d_in[7];
    const float* dt_b     = (const float*)d_in[8];
    const float* A_log    = (const float*)d_in[9];
    const float* Dp       = (const float*)d_in[10];
    const float* out_proj = (const float*)d_in[11];
    const float* W_cls    = (const float*)d_in[12];
    const float* b_cls    = (const float*)d_in[13];
    float* ws  = (float*)d_ws;
    float* out = (float*)d_out;

    precompute_kernel<<<354, 256, 0, stream>>>(W_in, b_in, in_proj, x_proj, dt_proj,
                                               out_proj, W_cls, A_log,
                                               ws + WS_WCIN, ws + WS_BCIN, ws + WS_WDT,
                                               ws + WS_WCOUT, ws + WS_ANEG);
    gemm_in_kernel<<<1024, 256, 0, stream>>>(input, ws + WS_WCIN, ws + WS_BCIN, ws + WS_XZ);
    conv_silu_kernel<<<NROWS, 256, 0, stream>>>(ws + WS_XZ, conv_w, conv_b, ws + WS_XC);
    gemm_xproj_kernel<<<576, 256, 0, stream>>>(ws + WS_XC, x_proj, ws + WS_WDT, dt_b,
                                               ws + WS_BC, ws + WS_DELTA);
    scan_pass1_kernel<<<dim3(SEG, B_), 256, 0, stream>>>(ws + WS_DELTA, ws + WS_XC, ws + WS_BC,
                                                         ws + WS_ANEG, ws + WS_P, ws + WS_Q);
    scan_mid_kernel<<<B_, 256, 0, stream>>>(ws + WS_P, ws + WS_Q, ws + WS_H0);
    scan_pass2_kernel<<<dim3(SEG, B_), 256, 0, stream>>>(ws + WS_XC, ws + WS_DELTA, ws + WS_BC,
                                                         ws + WS_ANEG, ws + WS_H0, Dp, ws + WS_XZ);
    gemm_out_kernel<<<64, 256, 0, stream>>>(ws + WS_XC, ws + WS_WCOUT, b_cls, out);
}